// ACM_44736379355568
// MI455X (gfx1250) — compile-verified
//
#include <hip/hip_runtime.h>
#include <math.h>

#define BATCH 32
#define HW 3136            // 56*56
#define CH 512
#define HEADS 32
#define CPH 16             // channels per head
#define BSTRIDE (HW * CH)      // 1605632 floats per batch of x
#define WMAP (HW * HEADS)      // 100352 floats per batch of a weight map
#define SLICE (CPH * HW)       // 50176 floats per (b,g) pooling slice

typedef __attribute__((ext_vector_type(2))) float v2f;
typedef __attribute__((ext_vector_type(8))) float v8f;

// ---- workspace layout (floats) ----
#define OFF_PART 0                        // 32*8*512     = 131072
#define OFF_MU   131072                   // 32*512       = 16384
#define OFF_CA   147456                   // 32*32        = 1024
#define OFF_CS   148480                   // 1024
#define OFF_MUL  149504                   // 16384
#define OFF_DIFF 165888                   // 16384
#define OFF_WA   182272                   // 32*100352    = 3211264
#define OFF_WS   3393536                  // 3211264  -> total 6604800 floats (~25.2MB)

// K1: one streaming pass over x.
//  - partial per-channel sums (for mu), 8 spatial chunks per batch
//  - raw grouped-conv dot maps for both attends (no mu correction yet)
__global__ void k1_mu_weights(const float* __restrict__ x,
                              const float* __restrict__ wa,
                              const float* __restrict__ wsv,
                              float* __restrict__ part,
                              float* __restrict__ wrA,
                              float* __restrict__ wrS) {
  __shared__ float waL[HEADS * CPH];
  __shared__ float wsL[HEADS * CPH];
  __shared__ float pbuf[8 * CH];   // [r][c]
  const int t = threadIdx.x;
  const int b = blockIdx.y;
  const int chunk = blockIdx.x;    // 0..7, 392 rows each
  for (int i = t; i < HEADS * CPH; i += 256) { waL[i] = wa[i]; wsL[i] = wsv[i]; }
  __syncthreads();

  const int g = t & 31;            // head (lane)
  const int r = t >> 5;            // 0..7 (wave id -> row phase)
  float acc[CPH];
#pragma unroll
  for (int i = 0; i < CPH; ++i) acc[i] = 0.0f;

  const float* xb = x + (size_t)b * BSTRIDE;
  for (int it = 0; it < 49; ++it) {
    const int s = chunk * 392 + r + it * 8;
    const float4* xp = (const float4*)(xb + (size_t)s * CH + g * CPH);
    float xr[16];
    *(float4*)(xr + 0)  = xp[0];
    *(float4*)(xr + 4)  = xp[1];
    *(float4*)(xr + 8)  = xp[2];
    *(float4*)(xr + 12) = xp[3];
    float dA = 0.0f, dS = 0.0f;
#pragma unroll
    for (int i = 0; i < CPH; ++i) {
      const float xi = xr[i];
      acc[i] += xi;
      dA += xi * waL[g * CPH + i];
      dS += xi * wsL[g * CPH + i];
    }
    const size_t widx = (size_t)b * WMAP + (size_t)s * HEADS + g;
    wrA[widx] = dA;
    wrS[widx] = dS;
  }
#pragma unroll
  for (int i = 0; i < CPH; ++i) pbuf[r * CH + g * CPH + i] = acc[i];
  __syncthreads();
  for (int c = t; c < CH; c += 256) {
    float s = 0.0f;
#pragma unroll
    for (int k = 0; k < 8; ++k) s += pbuf[k * CH + c];
    part[((size_t)b * 8 + chunk) * CH + c] = s;
  }
}

// K2: finalize mu; per-head softmax corrections (dot(mu,w)-bias); modulate MLP.
__global__ void k2_finalize(const float* __restrict__ wa,
                            const float* __restrict__ wsv,
                            const float* __restrict__ ba,
                            const float* __restrict__ bs,
                            const float* __restrict__ w1,
                            const float* __restrict__ b1,
                            const float* __restrict__ w2,
                            const float* __restrict__ b2,
                            const float* __restrict__ part,
                            float* __restrict__ mu,
                            float* __restrict__ corrA,
                            float* __restrict__ corrS,
                            float* __restrict__ mulv) {
  __shared__ float muL[CH];
  const int t = threadIdx.x;
  const int b = blockIdx.x;
  for (int c = t; c < CH; c += 256) {
    float s = 0.0f;
#pragma unroll
    for (int k = 0; k < 8; ++k) s += part[((size_t)b * 8 + k) * CH + c];
    s *= (1.0f / (float)HW);
    mu[(size_t)b * CH + c] = s;
    muL[c] = s;
  }
  __syncthreads();
  if (t < HEADS) {
    const int g = t;
    float cA = 0.0f, cS = 0.0f;
#pragma unroll
    for (int i = 0; i < CPH; ++i) {
      cA += muL[g * CPH + i] * wa[g * CPH + i];
      cS += muL[g * CPH + i] * wsv[g * CPH + i];
    }
    corrA[b * HEADS + g] = cA - ba[g];
    corrS[b * HEADS + g] = cS - bs[g];
    // modulate: 16 -> 8 (relu) -> 16 (sigmoid), per head
    float y1[8];
#pragma unroll
    for (int o = 0; o < 8; ++o) {
      float a = b1[g * 8 + o];
#pragma unroll
      for (int i = 0; i < CPH; ++i) a += muL[g * CPH + i] * w1[(g * CPH + i) * 8 + o];
      y1[o] = fmaxf(a, 0.0f);
    }
#pragma unroll
    for (int j = 0; j < CPH; ++j) {
      float a = b2[g * CPH + j];
#pragma unroll
      for (int o = 0; o < 8; ++o) a += y1[o] * w2[(g * 8 + o) * CPH + j];
      mulv[(size_t)b * CH + g * CPH + j] = 1.0f / (1.0f + expf(-a));
    }
  }
}

// K3: per (b, chunk g): chunk softmax for both maps, then WMMA f32 16x16x4 pooling.
// diff[b, g*16+c] = mus_add - mus_sub
__global__ void k3_attend(const float* __restrict__ x,
                          const float* __restrict__ wrA,
                          const float* __restrict__ wrS,
                          const float* __restrict__ mu,
                          const float* __restrict__ corrA,
                          const float* __restrict__ corrS,
                          float* __restrict__ diff) {
  __shared__ float ea[HW];
  __shared__ float es[HW];
  __shared__ float muL[CH];
  __shared__ float red[128];
  __shared__ float cAL[HEADS];
  __shared__ float cSL[HEADS];
  __shared__ float partA[64];
  __shared__ float partS[64];

  const int t = threadIdx.x;   // 128 threads = 4 waves
  const int g = blockIdx.x;    // flat chunk index 0..31
  const int b = blockIdx.y;

  for (int i = t; i < CH; i += 128) muL[i] = mu[(size_t)b * CH + i];
  if (t < HEADS) { cAL[t] = corrA[b * HEADS + t]; cSL[t] = corrS[b * HEADS + t]; }
  __syncthreads();

  // ---- softmax over contiguous 3136-element chunk, map A then map S ----
  const float* pA = wrA + (size_t)b * WMAP + (size_t)g * HW;
  const float* pS = wrS + (size_t)b * WMAP + (size_t)g * HW;

  float lm = -3.4e38f;
  for (int e = t; e < HW; e += 128) {
    const float v = pA[e] - cAL[e & 31];   // head of flat element e is e&31
    ea[e] = v;
    lm = fmaxf(lm, v);
  }
  red[t] = lm; __syncthreads();
  for (int off = 64; off > 0; off >>= 1) { if (t < off) red[t] = fmaxf(red[t], red[t + off]); __syncthreads(); }
  const float mA = red[0]; __syncthreads();
  float ls = 0.0f;
  for (int e = t; e < HW; e += 128) { const float w = expf(ea[e] - mA); ea[e] = w; ls += w; }
  red[t] = ls; __syncthreads();
  for (int off = 64; off > 0; off >>= 1) { if (t < off) red[t] += red[t + off]; __syncthreads(); }
  const float invDA = 1.0f / red[0]; __syncthreads();

  lm = -3.4e38f;
  for (int e = t; e < HW; e += 128) {
    const float v = pS[e] - cSL[e & 31];
    es[e] = v;
    lm = fmaxf(lm, v);
  }
  red[t] = lm; __syncthreads();
  for (int off = 64; off > 0; off >>= 1) { if (t < off) red[t] = fmaxf(red[t], red[t + off]); __syncthreads(); }
  const float mS = red[0]; __syncthreads();
  ls = 0.0f;
  for (int e = t; e < HW; e += 128) { const float w = expf(es[e] - mS); es[e] = w; ls += w; }
  red[t] = ls; __syncthreads();
  for (int off = 64; off > 0; off >>= 1) { if (t < off) red[t] += red[t + off]; __syncthreads(); }
  const float invDS = 1.0f / red[0]; __syncthreads();

  // ---- WMMA pooling: D[16c x 16] += A[16c x 4s] * B[4s x 16(broadcast)] ----
  // f32 16x16x4 A layout (ISA 7.12.2): lane 0-15 -> M=lane, {VGPR0,VGPR1}={K0,K1};
  // lane 16-31 -> M=lane-16, {K2,K3}. B uses the same K-half split; we broadcast
  // the weight over N so every column of D holds the same dot product.
  const int wave = t >> 5, lane = t & 31;
  const int c = lane & 15, hi = lane >> 4, k0 = hi * 2;
  const float* xs = x + (size_t)b * BSTRIDE + (size_t)g * SLICE;
  v8f accA = {};
  v8f accS = {};
  const int s_begin = wave * 784, s_end = s_begin + 784;
  for (int s0 = s_begin; s0 < s_end; s0 += 4) {
    const int idx = c * HW + s0 + k0;
    const v2f xv = *(const v2f*)(xs + idx);
    const int chn = (c * 64 + s0 + k0) & 511;   // true channel of flat element (even)
    v2f a; a.x = xv.x - muL[chn]; a.y = xv.y - muL[chn + 1];
    v2f bA; bA.x = ea[s0 + k0]; bA.y = ea[s0 + k0 + 1];
    v2f bS; bS.x = es[s0 + k0]; bS.y = es[s0 + k0 + 1];
    accA = __builtin_amdgcn_wmma_f32_16x16x4_f32(false, a, false, bA, (short)0, accA, false, false);
    accS = __builtin_amdgcn_wmma_f32_16x16x4_f32(false, a, false, bS, (short)0, accS, false, false);
    if ((s0 & 127) == 0) __builtin_prefetch(xs + idx + 512, 0, 0);
  }
  // D layout: VGPR r, lanes 0-15 -> M=r (N=lane), lanes 16-31 -> M=8+r.
  if (lane == 0) {
#pragma unroll
    for (int r = 0; r < 8; ++r) { partA[wave * 16 + r] = accA[r]; partS[wave * 16 + r] = accS[r]; }
  }
  if (lane == 16) {
#pragma unroll
    for (int r = 0; r < 8; ++r) { partA[wave * 16 + 8 + r] = accA[r]; partS[wave * 16 + 8 + r] = accS[r]; }
  }
  __syncthreads();
  if (t < CPH) {
    float sA = 0.0f, sS = 0.0f;
#pragma unroll
    for (int w = 0; w < 4; ++w) { sA += partA[w * 16 + t]; sS += partS[w * 16 + t]; }
    diff[(size_t)b * CH + g * CPH + t] = sA * invDA - sS * invDS;
  }
}

// K4: out = (x + diff[b,c]) * mul[b,c], float4 streams.
__global__ void k4_out(const float* __restrict__ x,
                       const float* __restrict__ diff,
                       const float* __restrict__ mulv,
                       float* __restrict__ out) {
  __shared__ float dL[CH];
  __shared__ float mL[CH];
  const int t = threadIdx.x;
  const int b = blockIdx.y;
  for (int i = t; i < CH; i += 256) { dL[i] = diff[(size_t)b * CH + i]; mL[i] = mulv[(size_t)b * CH + i]; }
  __syncthreads();
  const size_t j = ((size_t)blockIdx.x * 256 + t) * 4;
  const float4 xv = *(const float4*)(x + (size_t)b * BSTRIDE + j);
  const int ch = (int)(j & 511);
  float4 o;
  o.x = (xv.x + dL[ch + 0]) * mL[ch + 0];
  o.y = (xv.y + dL[ch + 1]) * mL[ch + 1];
  o.z = (xv.z + dL[ch + 2]) * mL[ch + 2];
  o.w = (xv.w + dL[ch + 3]) * mL[ch + 3];
  *(float4*)(out + (size_t)b * BSTRIDE + j) = o;
}

extern "C" void kernel_launch(void* const* d_in, const int* in_sizes, int n_in,
                              void* d_out, int out_size, void* d_ws, size_t ws_size,
                              hipStream_t stream) {
  const float* x  = (const float*)d_in[0];
  const float* wa = (const float*)d_in[1];
  const float* ba = (const float*)d_in[2];
  const float* wsv= (const float*)d_in[3];
  const float* bs = (const float*)d_in[4];
  const float* w1 = (const float*)d_in[5];
  const float* b1 = (const float*)d_in[6];
  const float* w2 = (const float*)d_in[7];
  const float* b2 = (const float*)d_in[8];
  float* out = (float*)d_out;
  float* W = (float*)d_ws;

  float* part  = W + OFF_PART;
  float* mu    = W + OFF_MU;
  float* corrA = W + OFF_CA;
  float* corrS = W + OFF_CS;
  float* mulv  = W + OFF_MUL;
  float* diff  = W + OFF_DIFF;
  float* wrA   = W + OFF_WA;
  float* wrS   = W + OFF_WS;

  k1_mu_weights<<<dim3(8, BATCH), 256, 0, stream>>>(x, wa, wsv, part, wrA, wrS);
  k2_finalize<<<BATCH, 256, 0, stream>>>(wa, wsv, ba, bs, w1, b1, w2, b2,
                                         part, mu, corrA, corrS, mulv);
  k3_attend<<<dim3(HEADS, BATCH), 128, 0, stream>>>(x, wrA, wrS, mu, corrA, corrS, diff);
  k4_out<<<dim3(BSTRIDE / (256 * 4), BATCH), 256, 0, stream>>>(x, diff, mulv, out);
}